// MyGCN_50457275793466
// MI455X (gfx1250) — compile-verified
//
#include <hip/hip_runtime.h>
#include <math.h>

// GCN: N nodes, E edges, 512 -> 16 -> 40.
// ws layout: [deg/dinv : N f32][h : N*16 f32][agg : N*16 f32][norm : E f32 (optional)]

typedef __attribute__((ext_vector_type(2))) float v2f;
typedef __attribute__((ext_vector_type(8))) float v8f;

// ---------------- degree ----------------
__global__ void gcn_deg_kernel(const int* __restrict__ dst,
                               const float* __restrict__ ew,
                               float* __restrict__ deg, int e) {
    int i = blockIdx.x * blockDim.x + threadIdx.x;
    if (i < e) atomicAdd(&deg[dst[i]], ew[i]);
}

// ---------------- dinv = 1/sqrt(deg+1), in place ----------------
__global__ void gcn_dinv_kernel(float* __restrict__ deg, int nn) {
    int i = blockIdx.x * blockDim.x + threadIdx.x;
    if (i < nn) {
        float d = deg[i] + 1.0f;            // self-loop weight 1
        deg[i] = 1.0f / sqrtf(d);           // deg >= 1 always
    }
}

// ---------------- norm[e] = dinv[src]*ew*dinv[dst] (shared by both scatters) ---
__global__ void gcn_norm_kernel(const int* __restrict__ src,
                                const int* __restrict__ dst,
                                const float* __restrict__ ew,
                                const float* __restrict__ dinv,
                                float* __restrict__ norm, int e) {
    int i = blockIdx.x * blockDim.x + threadIdx.x;
    if (i < e) norm[i] = dinv[src[i]] * ew[i] * dinv[dst[i]];
}

// ---------------- GEMM1: h = x @ W1 via fp32 WMMA ----------------
// One wave per 16-row tile. K=512 consumed 8 at a time (2 WMMAs / float4 load).
// K-permutation: step m, lanes 0-15 use x cols 8m..8m+3, lanes 16-31 use 8m+4..8m+7;
// B fragments of W1 are loaded to match the same permutation.
__global__ void gcn_gemm1_wmma(const float* __restrict__ x,
                               const float* __restrict__ w1,
                               float* __restrict__ h, int nn) {
    int wavesPerBlock = blockDim.x >> 5;
    int waveId = blockIdx.x * wavesPerBlock + ((int)threadIdx.x >> 5);
    int rb = waveId * 16;
    if (rb >= nn) return;

    int lane = threadIdx.x & 31;
    int m    = lane & 15;       // A: row-in-tile / B,C,D: column
    int half = lane >> 4;

    int arow = rb + m;
    if (arow >= nn) arow = nn - 1;          // clamp (duplicate loads, guarded stores)
    const float* xrow = x + (size_t)arow * 512 + half * 4;

    v8f acc = {};
    #pragma unroll 4
    for (int km = 0; km < 64; ++km) {
        float4 a4 = *(const float4*)(xrow + km * 8);
        int k0 = km * 8 + half * 4;         // this lane-half's first x column
        v2f a, b;
        // step A: K slots {0,1} (lanes 0-15) / {2,3} (lanes 16-31) -> cols k0,k0+1
        a.x = a4.x; a.y = a4.y;
        b.x = w1[(k0 + 0) * 16 + m];
        b.y = w1[(k0 + 1) * 16 + m];
        acc = __builtin_amdgcn_wmma_f32_16x16x4_f32(false, a, false, b,
                                                    (short)0, acc, false, false);
        // step B: cols k0+2, k0+3
        a.x = a4.z; a.y = a4.w;
        b.x = w1[(k0 + 2) * 16 + m];
        b.y = w1[(k0 + 3) * 16 + m];
        acc = __builtin_amdgcn_wmma_f32_16x16x4_f32(false, a, false, b,
                                                    (short)0, acc, false, false);
    }
    // C/D layout: VGPR r -> row (r + 8*half), col m
    if (rb + 16 <= nn) {                    // fast path: full tile, no guards
        float* hp = h + (size_t)(rb + 8 * half) * 16 + m;
        #pragma unroll
        for (int r = 0; r < 8; ++r) hp[(size_t)r * 16] = acc[r];
    } else {
        #pragma unroll
        for (int r = 0; r < 8; ++r) {
            int row = rb + r + 8 * half;
            if (row < nn) h[(size_t)row * 16 + m] = acc[r];
        }
    }
}

// ---------------- edge scatter: agg[dst] += norm * h[src] ----------------
// 4 threads per edge, float4 of hidden dims each (b128 gather, 4x f32 atomics).
__global__ void gcn_scatter4_kernel(const int* __restrict__ src,
                                    const int* __restrict__ dst,
                                    const float* __restrict__ norm,
                                    const float* __restrict__ h,
                                    float* __restrict__ agg, int e) {
    unsigned int t = blockIdx.x * blockDim.x + threadIdx.x;
    int eid = (int)(t >> 2);
    int q   = (int)(t & 3) * 4;
    if (eid < e) {
        int s  = src[eid];
        int tt = dst[eid];
        float nrm = norm[eid];
        float4 hv = *(const float4*)(h + (size_t)s * 16 + q);
        float* ap = agg + (size_t)tt * 16 + q;
        atomicAdd(ap + 0, hv.x * nrm);
        atomicAdd(ap + 1, hv.y * nrm);
        atomicAdd(ap + 2, hv.z * nrm);
        atomicAdd(ap + 3, hv.w * nrm);
    }
}

// fallback when ws can't hold norm[]: compute norm inline
__global__ void gcn_scatter4_inline_kernel(const int* __restrict__ src,
                                           const int* __restrict__ dst,
                                           const float* __restrict__ ew,
                                           const float* __restrict__ dinv,
                                           const float* __restrict__ h,
                                           float* __restrict__ agg, int e) {
    unsigned int t = blockIdx.x * blockDim.x + threadIdx.x;
    int eid = (int)(t >> 2);
    int q   = (int)(t & 3) * 4;
    if (eid < e) {
        int s  = src[eid];
        int tt = dst[eid];
        float nrm = dinv[s] * ew[eid] * dinv[tt];
        float4 hv = *(const float4*)(h + (size_t)s * 16 + q);
        float* ap = agg + (size_t)tt * 16 + q;
        atomicAdd(ap + 0, hv.x * nrm);
        atomicAdd(ap + 1, hv.y * nrm);
        atomicAdd(ap + 2, hv.z * nrm);
        atomicAdd(ap + 3, hv.w * nrm);
    }
}

// ---------------- self-loop + bias + ReLU, in place on h (float4) ----------------
__global__ void gcn_selfloop_relu(const float* __restrict__ agg,
                                  float* __restrict__ h,
                                  const float* __restrict__ dinv,
                                  const float* __restrict__ b1, int nn) {
    int t = blockIdx.x * blockDim.x + threadIdx.x;
    if (t < nn * 4) {
        int j = t >> 2, q = (t & 3) * 4;
        float di  = dinv[j];
        float di2 = di * di;
        float4 av = *(const float4*)(agg + (size_t)j * 16 + q);
        float4 hv = *(const float4*)(h   + (size_t)j * 16 + q);
        float4 bv = *(const float4*)(b1 + q);
        float4 r;
        r.x = fmaxf(av.x + hv.x * di2 + bv.x, 0.0f);
        r.y = fmaxf(av.y + hv.y * di2 + bv.y, 0.0f);
        r.z = fmaxf(av.z + hv.z * di2 + bv.z, 0.0f);
        r.w = fmaxf(av.w + hv.w * di2 + bv.w, 0.0f);
        *(float4*)(h + (size_t)j * 16 + q) = r;
    }
}

// ---------------- final: z = agg2 + dinv^2*h2; out = log_softmax(z@W2 + b2) ----
__global__ void gcn_final_kernel(const float* __restrict__ agg,
                                 const float* __restrict__ h,
                                 const float* __restrict__ dinv,
                                 const float* __restrict__ w2,
                                 const float* __restrict__ b2,
                                 float* __restrict__ out, int nn) {
    int j = blockIdx.x * blockDim.x + threadIdx.x;
    if (j >= nn) return;
    float di = dinv[j];
    float di2 = di * di;
    float z[16];
    #pragma unroll
    for (int k = 0; k < 16; ++k)
        z[k] = agg[(size_t)j * 16 + k] + h[(size_t)j * 16 + k] * di2;

    float o[40];
    #pragma unroll
    for (int c = 0; c < 40; ++c) o[c] = b2[c];
    #pragma unroll
    for (int k = 0; k < 16; ++k) {
        float zk = z[k];
        #pragma unroll
        for (int c = 0; c < 40; ++c) o[c] += zk * w2[k * 40 + c];
    }
    float mx = o[0];
    #pragma unroll
    for (int c = 1; c < 40; ++c) mx = fmaxf(mx, o[c]);
    float s = 0.0f;
    #pragma unroll
    for (int c = 0; c < 40; ++c) s += expf(o[c] - mx);
    float l = logf(s);
    #pragma unroll
    for (int c = 0; c < 40; ++c) out[(size_t)j * 40 + c] = o[c] - mx - l;
}

extern "C" void kernel_launch(void* const* d_in, const int* in_sizes, int n_in,
                              void* d_out, int out_size, void* d_ws, size_t ws_size,
                              hipStream_t stream) {
    const float* x   = (const float*)d_in[0];
    const int*   src = (const int*)  d_in[1];
    const int*   dst = (const int*)  d_in[2];
    const float* ew  = (const float*)d_in[3];
    const float* W1  = (const float*)d_in[4];
    const float* b1  = (const float*)d_in[5];
    const float* W2  = (const float*)d_in[6];
    const float* b2  = (const float*)d_in[7];
    float* out = (float*)d_out;

    int nn = in_sizes[0] / 512;     // 100000
    int e  = in_sizes[1];           // 3200000

    // workspace carve-up (dinv overwrites deg in place)
    size_t degElems = ((size_t)nn + 63) & ~(size_t)63;
    float* deg  = (float*)d_ws;
    float* h    = deg + degElems;
    float* agg  = h + (size_t)nn * 16;
    float* norm = agg + (size_t)nn * 16;
    size_t baseBytes = (degElems + (size_t)nn * 32) * sizeof(float);
    bool haveNorm = ws_size >= baseBytes + (size_t)e * sizeof(float);

    hipMemsetAsync(deg, 0, sizeof(float) * (size_t)nn, stream);
    hipMemsetAsync(agg, 0, sizeof(float) * (size_t)nn * 16, stream);

    gcn_deg_kernel<<<(e + 255) / 256, 256, 0, stream>>>(dst, ew, deg, e);
    gcn_dinv_kernel<<<(nn + 255) / 256, 256, 0, stream>>>(deg, nn);
    if (haveNorm)
        gcn_norm_kernel<<<(e + 255) / 256, 256, 0, stream>>>(src, dst, ew, deg, norm, e);

    int tiles = (nn + 15) / 16;                 // one wave per tile
    gcn_gemm1_wmma<<<(tiles + 3) / 4, 128, 0, stream>>>(x, W1, h, nn);

    unsigned int st = (unsigned int)e * 4u;
    if (haveNorm)
        gcn_scatter4_kernel<<<(st + 255) / 256, 256, 0, stream>>>(src, dst, norm, h, agg, e);
    else
        gcn_scatter4_inline_kernel<<<(st + 255) / 256, 256, 0, stream>>>(src, dst, ew, deg, h, agg, e);

    gcn_selfloop_relu<<<(nn * 4 + 255) / 256, 256, 0, stream>>>(agg, h, deg, b1, nn);

    hipMemsetAsync(agg, 0, sizeof(float) * (size_t)nn * 16, stream);
    if (haveNorm)
        gcn_scatter4_kernel<<<(st + 255) / 256, 256, 0, stream>>>(src, dst, norm, h, agg, e);
    else
        gcn_scatter4_inline_kernel<<<(st + 255) / 256, 256, 0, stream>>>(src, dst, ew, deg, h, agg, e);

    gcn_final_kernel<<<(nn + 255) / 256, 256, 0, stream>>>(agg, h, deg, W2, b2, out, nn);
}